// GIBGNN_59863254171699
// MI455X (gfx1250) — compile-verified
//
#include <hip/hip_runtime.h>
#include <hip/hip_bf16.h>

typedef __attribute__((ext_vector_type(2))) float v2f;
typedef __attribute__((ext_vector_type(8))) float v8f;

#define DDIM 128
#define BN_EPS 1e-5f
#define APAD 132   // 128 + 4 pad -> conflict-free LDS b64 reads

// -------------------------------------------------------------------------
// GEMM: C = A(N x 128) @ B(128 x 128) + bias, plus per-column sum/sumsq
// accumulated via atomics for the following BatchNorm.
// Block = 256 threads (8 wave32); block computes rows [16*bx, 16*bx+16),
// wave w computes cols [16w, 16w+16). A tile staged in LDS so the block
// reads its 16 rows completely before writing them -> in-place safe.
// Uses fp32 WMMA (16x16x4) to match the fp32 reference exactly.
// -------------------------------------------------------------------------
__global__ __launch_bounds__(256)
void k_gemm_bn(const float* __restrict__ A, const float* __restrict__ B,
               const float* __restrict__ bias, float* __restrict__ C,
               float* __restrict__ stat_sum, float* __restrict__ stat_sq)
{
    __shared__ float As[16 * APAD];
    const int tid  = threadIdx.x;
    const int row0 = blockIdx.x * 16;

    // Stage A tile: 2048 floats = 512 float4, 2 per thread, coalesced.
    #pragma unroll
    for (int it = 0; it < 2; ++it) {
        int v4 = tid + it * 256;          // float4 index 0..511
        int r  = v4 >> 5;                 // 32 float4 per row
        int cc = (v4 & 31) << 2;
        float4 src = *(const float4*)(A + (size_t)(row0 + r) * DDIM + cc);
        *(float4*)(&As[r * APAD + cc]) = src;
    }
    __syncthreads();

    const int lane = tid & 31;
    const int wave = tid >> 5;
    const int half = lane >> 4;           // 0: K pair {0,1}/rows 0-7, 1: {2,3}/rows 8-15
    const int mn   = lane & 15;           // A row (operand) / C column (lane)
    const int col  = wave * 16 + mn;

    v8f acc = {0.f, 0.f, 0.f, 0.f, 0.f, 0.f, 0.f, 0.f};

    const float* Ap = &As[mn * APAD + half * 2];  // A[mn][2*half + k]
    const float* Bp = B + col;                    // B[k][col] at Bp[k*128]

    #pragma unroll
    for (int s = 0; s < 32; ++s) {
        const int k = s * 4;
        v2f a = *(const v2f*)(Ap + k);                 // A[mn][k+2h], A[mn][k+2h+1]
        v2f b;
        b.x = Bp[(k + 2 * half)     * DDIM];           // B[k+2h][col]
        b.y = Bp[(k + 2 * half + 1) * DDIM];           // B[k+2h+1][col]
        acc = __builtin_amdgcn_wmma_f32_16x16x4_f32(
                  false, a, false, b, (short)0, acc, false, false);
    }

    const float bc = bias[col];
    float s1 = 0.f, s2 = 0.f;
    float* Crow = C + (size_t)(row0 + half * 8) * DDIM + col;
    #pragma unroll
    for (int j = 0; j < 8; ++j) {                      // VGPR j -> row j (+8 if half)
        float v = acc[j] + bc;
        Crow[(size_t)j * DDIM] = v;
        s1 += v;
        s2 += v * v;
    }
    atomicAdd(&stat_sum[col], s1);
    atomicAdd(&stat_sq[col],  s2);
}

// h[dst] += x[src] : one wave per edge, 4 floats per lane. L2-resident atomics.
__global__ __launch_bounds__(256)
void k_scatter(const float* __restrict__ xsrc, const int* __restrict__ src,
               const int* __restrict__ dst, float* __restrict__ h, int E)
{
    const int gid = blockIdx.x * 256 + threadIdx.x;
    const int e = gid >> 5;
    if (e >= E) return;
    const int lane = gid & 31;
    const int s = src[e], d = dst[e];
    const float4 v = *(const float4*)(xsrc + (size_t)s * DDIM + lane * 4);
    float* hp = h + (size_t)d * DDIM + lane * 4;
    atomicAdd(hp + 0, v.x);
    atomicAdd(hp + 1, v.y);
    atomicAdd(hp + 2, v.z);
    atomicAdd(hp + 3, v.w);
}

// Layer-0 init: h = x ; block 0 also zeroes the 512-float stats region.
__global__ __launch_bounds__(256)
void k_init(const float* __restrict__ x, float* __restrict__ h,
            float* __restrict__ stats, long total4)
{
    if (blockIdx.x == 0) { stats[threadIdx.x] = 0.f; stats[threadIdx.x + 256] = 0.f; }
    const long gid = (long)blockIdx.x * 256 + threadIdx.x;
    if (gid < total4) ((float4*)h)[gid] = ((const float4*)x)[gid];
}

// x = relu(A*(t2 - M) + ob), elementwise (in-place capable). Optionally also
// initializes the next layer's h and zeroes the stats region for its GEMMs.
__global__ __launch_bounds__(256)
void k_activate(const float* __restrict__ t2, const float* __restrict__ Ac,
                const float* __restrict__ Mc, const float* __restrict__ ob,
                float* __restrict__ xout, float* __restrict__ hinit,
                float* __restrict__ stats, long total4)
{
    if (stats && blockIdx.x == 0) { stats[threadIdx.x] = 0.f; stats[threadIdx.x + 256] = 0.f; }
    const long gid = (long)blockIdx.x * 256 + threadIdx.x;
    if (gid >= total4) return;
    const int c0 = (int)((gid & 31) << 2);     // (gid*4) % 128
    float4 t = ((const float4*)t2)[gid];
    float4 r;
    r.x = fmaxf(Ac[c0 + 0] * (t.x - Mc[c0 + 0]) + ob[c0 + 0], 0.f);
    r.y = fmaxf(Ac[c0 + 1] * (t.y - Mc[c0 + 1]) + ob[c0 + 1], 0.f);
    r.z = fmaxf(Ac[c0 + 2] * (t.z - Mc[c0 + 2]) + ob[c0 + 2], 0.f);
    r.w = fmaxf(Ac[c0 + 3] * (t.w - Mc[c0 + 3]) + ob[c0 + 3], 0.f);
    ((float4*)xout)[gid] = r;
    if (hinit) ((float4*)hinit)[gid] = r;
}

// Fold BN1 into GEMM2: W2eff[k,:] = s1[k]*W2[k,:];  b2eff = b2 + shift1^T W2.
__global__ __launch_bounds__(128)
void k_fold(const float* __restrict__ sum1, const float* __restrict__ sq1,
            const float* __restrict__ g1, const float* __restrict__ bt1,
            const float* __restrict__ W2, const float* __restrict__ b2,
            float* __restrict__ W2eff, float* __restrict__ b2eff, float invN)
{
    __shared__ float sc[DDIM], sh[DDIM];
    const int c = threadIdx.x;
    const float m = sum1[c] * invN;
    const float v = sq1[c] * invN - m * m;
    const float s = g1[c] * rsqrtf(v + BN_EPS);
    sc[c] = s;
    sh[c] = bt1[c] - m * s;
    __syncthreads();
    float acc = b2[c];
    for (int k = 0; k < DDIM; ++k) {
        const float w = W2[k * DDIM + c];
        W2eff[k * DDIM + c] = sc[k] * w;
        acc += sh[k] * w;
    }
    b2eff[c] = acc;
}

// BN2 composed with outer BN + ReLU:  x = relu(A*(t2-m2)+ob); bt2 cancels.
__global__ __launch_bounds__(128)
void k_outer(const float* __restrict__ sum2, const float* __restrict__ sq2,
             const float* __restrict__ g2, const float* __restrict__ og,
             float* __restrict__ Ac, float* __restrict__ Mc, float invN)
{
    const int c = threadIdx.x;
    const float m  = sum2[c] * invN;
    const float v  = sq2[c] * invN - m * m;
    const float a2 = g2[c] * rsqrtf(v + BN_EPS);
    Ac[c] = og[c] * a2 * rsqrtf(a2 * a2 * v + BN_EPS);
    Mc[c] = m;
}

__global__ __launch_bounds__(256)
void k_zero(float* __restrict__ p, int n)
{
    const int i = blockIdx.x * 256 + threadIdx.x;
    if (i < n) p[i] = 0.f;
}

// ge[batch[i]] += x[i] * w[i] : one wave per node, 4 floats per lane.
__global__ __launch_bounds__(256)
void k_pool(const float* __restrict__ xe, const float* __restrict__ nw,
            const int* __restrict__ batch, float* __restrict__ ge, int N)
{
    const int gid = blockIdx.x * 256 + threadIdx.x;
    const int i = gid >> 5;
    if (i >= N) return;
    const int lane = gid & 31;
    const float w = nw[i];
    const int g = batch[i];
    const float4 v = *(const float4*)(xe + (size_t)i * DDIM + lane * 4);
    float* gp = ge + (size_t)g * DDIM + lane * 4;
    atomicAdd(gp + 0, v.x * w);
    atomicAdd(gp + 1, v.y * w);
    atomicAdd(gp + 2, v.z * w);
    atomicAdd(gp + 3, v.w * w);
}

// logits = ge @ fcW + fcb  (128x128 @ 128x2, one block)
__global__ __launch_bounds__(256)
void k_fc(const float* __restrict__ ge, const float* __restrict__ fcW,
          const float* __restrict__ fcb, float* __restrict__ outp, int G, int Cc)
{
    const int t = blockIdx.x * 256 + threadIdx.x;
    if (t >= G * Cc) return;
    const int g = t / Cc, c = t % Cc;
    float acc = fcb[c];
    for (int d = 0; d < DDIM; ++d) acc += ge[(size_t)g * DDIM + d] * fcW[d * Cc + c];
    outp[t] = acc;
}

extern "C" void kernel_launch(void* const* d_in, const int* in_sizes, int n_in,
                              void* d_out, int out_size, void* d_ws, size_t ws_size,
                              hipStream_t stream)
{
    const float* x    = (const float*)d_in[0];
    const int*   ei   = (const int*)  d_in[1];
    const float* nw   = (const float*)d_in[2];
    const int*   bat  = (const int*)  d_in[3];
    const float* W1   = (const float*)d_in[4];
    const float* b1   = (const float*)d_in[5];
    const float* g1   = (const float*)d_in[6];
    const float* bt1  = (const float*)d_in[7];
    const float* W2   = (const float*)d_in[8];
    const float* b2   = (const float*)d_in[9];
    const float* g2   = (const float*)d_in[10];
    /* bt2 = d_in[11] cancels analytically under the outer BN */
    const float* og   = (const float*)d_in[12];
    const float* ob   = (const float*)d_in[13];
    const float* fcW  = (const float*)d_in[14];
    const float* fcb  = (const float*)d_in[15];

    const int D = DDIM;
    const int N = in_sizes[0] / D;          // 100000 (divisible by 16)
    const int E = in_sizes[1] / 2;          // 1600000
    const int L = in_sizes[4] / (D * D);    // 3
    const int C = in_sizes[14] / D;         // 2
    const int G = (out_size - N * D) / (D + C);  // 128

    // Workspace layout (floats): ~102.5 MB total
    float* ws    = (float*)d_ws;
    float* P0    = ws;                          // N*D : xl -> t1 -> t2 (in-place)
    float* Hbuf  = P0 + (size_t)N * D;          // N*D : h (scatter target)
    float* W2eff = Hbuf + (size_t)N * D;        // D*D
    float* b2eff = W2eff + (size_t)D * D;       // D
    float* stats = b2eff + D;                   // 4*D: sum1, sq1, sum2, sq2
    float* Ac    = stats + 4 * D;               // D
    float* Mc    = Ac + D;                      // D

    float* node_emb = (float*)d_out;            // N*D
    float* ge       = node_emb + (size_t)N * D; // G*D
    float* logits   = ge + (size_t)G * D;       // G*C

    const int* srcIdx = ei;
    const int* dstIdx = ei + E;

    const long total4   = (long)N * D / 4;
    const int  nBlocks4 = (int)((total4 + 255) / 256);
    const int  eBlocks  = (int)(((long)E * 32 + 255) / 256);
    const int  gemmBlks = N / 16;
    const float invN    = 1.0f / (float)N;

    for (int l = 0; l < L; ++l) {
        if (l == 0) {
            k_init<<<nBlocks4, 256, 0, stream>>>(x, Hbuf, stats, total4);
        } else {
            // previous t2 sits in P0; activate in place + seed h + zero stats
            k_activate<<<nBlocks4, 256, 0, stream>>>(
                P0, Ac, Mc, ob + (size_t)(l - 1) * D, P0, Hbuf, stats, total4);
        }
        const float* xl = (l == 0) ? x : P0;

        // h[dst] += x[src]   (h pre-seeded with x -> GIN (1+eps)*x_i with eps=0)
        k_scatter<<<eBlocks, 256, 0, stream>>>(xl, srcIdx, dstIdx, Hbuf, E);

        // t1 = h @ W1[l] + b1[l]   (+ column stats for BN1)
        k_gemm_bn<<<gemmBlks, 256, 0, stream>>>(
            Hbuf, W1 + (size_t)l * D * D, b1 + (size_t)l * D,
            P0, stats, stats + D);

        // fold BN1 into W2
        k_fold<<<1, 128, 0, stream>>>(
            stats, stats + D, g1 + (size_t)l * D, bt1 + (size_t)l * D,
            W2 + (size_t)l * D * D, b2 + (size_t)l * D, W2eff, b2eff, invN);

        // t2 = t1 @ W2eff + b2eff  (in-place over P0; + column stats for BN2)
        k_gemm_bn<<<gemmBlks, 256, 0, stream>>>(
            P0, W2eff, b2eff, P0, stats + 2 * D, stats + 3 * D);

        // collapse BN2 + outer BN into per-column affine (Ac, Mc)
        k_outer<<<1, 128, 0, stream>>>(
            stats + 2 * D, stats + 3 * D,
            g2 + (size_t)l * D, og + (size_t)l * D, Ac, Mc, invN);
    }

    // final activation -> node embeddings (d_out)
    k_activate<<<nBlocks4, 256, 0, stream>>>(
        P0, Ac, Mc, ob + (size_t)(L - 1) * D, node_emb, nullptr, nullptr, total4);

    // weighted graph pooling
    k_zero<<<(G * D + 255) / 256, 256, 0, stream>>>(ge, G * D);
    k_pool<<<(int)(((long)N * 32 + 255) / 256), 256, 0, stream>>>(node_emb, nw, bat, ge, N);

    // classifier
    k_fc<<<(G * C + 255) / 256, 256, 0, stream>>>(ge, fcW, fcb, logits, G, C);
}